// PSRoIAlignPool_40252433498790
// MI455X (gfx1250) — compile-verified
//
#include <hip/hip_runtime.h>
#include <hip/hip_bf16.h>

// PS-RoI-Align for MI455X (gfx1250).
// Gather-bound, L2-resident workload (features = 128MB < 192MB L2). No GEMM
// structure exists (contraction operands co-vary with output indices), so no
// WMMA; instead we use the CDNA5 async global->LDS path for ROI staging and
// global_prefetch for the feature rows.

#define POOLED     7
#define ODIM       10
#define GS         7
#define CDIM       (ODIM * GS * GS)     // 490
#define HDIM       128
#define WDIM       128
#define PLANE      (HDIM * WDIM)        // 16384
#define PER_ROI    (ODIM * POOLED * POOLED) // 490
#define SCALE      0.0625f

#if __has_builtin(__builtin_amdgcn_global_load_async_to_lds_b32) && \
    __has_builtin(__builtin_amdgcn_s_wait_asynccnt)
#define USE_ASYNC_LDS 1
#else
#define USE_ASYNC_LDS 0
#endif

typedef __attribute__((address_space(1))) int global_i32;
typedef __attribute__((address_space(3))) int lds_i32;

__device__ __forceinline__ int imin(int a, int b) { return a < b ? a : b; }

__global__ __launch_bounds__(256) void psroi_align_kernel(
    const float* __restrict__ features,
    const float* __restrict__ rois,
    float* __restrict__ out,
    int total, int R)
{
    __shared__ float lds_roi[16];

    const int tid  = threadIdx.x;
    const int base = blockIdx.x * 256;
    const int rBeg = base / PER_ROI;          // block spans <= 2 ROIs (256 < 490)
    const int nfl  = (R - rBeg) >= 2 ? 10 : 5; // floats of roi table to stage

    if (tid < nfl) {
#if USE_ASYNC_LDS
        // CDNA5 async global->LDS copy (ASYNCcnt): per-lane 4B transfer.
        __builtin_amdgcn_global_load_async_to_lds_b32(
            (global_i32*)const_cast<float*>(rois + rBeg * 5 + tid),
            (lds_i32*)&lds_roi[tid],
            /*offset=*/0, /*cpol=*/0);
#else
        lds_roi[tid] = rois[rBeg * 5 + tid];
#endif
    }
#if USE_ASYNC_LDS
    __builtin_amdgcn_s_wait_asynccnt(0);
#endif
    __syncthreads();

    const int idx = base + tid;
    if (idx >= total) return;

    // idx = ((r*10 + c)*7 + ph)*7 + pw   (coalesced stores)
    const int r   = idx / PER_ROI;
    const int j   = idx - r * PER_ROI;
    const int c   = j / (POOLED * POOLED);
    const int bin = j - c * (POOLED * POOLED);
    const int ph  = bin / POOLED;
    const int pw  = bin - ph * POOLED;

    const float* roi = &lds_roi[(r - rBeg) * 5];
    const int   b  = (int)roi[0];
    const float x1 = roi[1] * SCALE;
    const float y1 = roi[2] * SCALE;
    const float x2 = roi[3] * SCALE;
    const float y2 = roi[4] * SCALE;

    const float roi_w = fmaxf(x2 - x1, 0.1f);
    const float roi_h = fmaxf(y2 - y1, 0.1f);
    const float bin_w = roi_w / 7.0f;   // match reference: divide, not *(1/7)
    const float bin_h = roi_h / 7.0f;

    // position-sensitive channel: c_in = (c*7 + ph)*7 + pw
    const int c_in = (c * GS + ph) * GS + pw;
    const float* plane = features + ((size_t)b * CDIM + (size_t)c_in) * PLANE;

    // Prefetch the sy=1 sample row while sy=0 is being gathered.
    {
        float ypf = fminf(fmaxf(y1 + ((float)ph + 0.75f) * bin_h, 0.0f), 127.0f);
        float xpf = fminf(fmaxf(x1 + ((float)pw + 0.25f) * bin_w, 0.0f), 127.0f);
        __builtin_prefetch(plane + (int)ypf * WDIM + (int)xpf, 0, 1);
    }

    float acc = 0.0f;
#pragma unroll
    for (int sy = 0; sy < 2; ++sy) {
        float yy = y1 + ((float)ph + (sy ? 0.75f : 0.25f)) * bin_h;
        yy = fminf(fmaxf(yy, 0.0f), 127.0f);
        const float y0f = floorf(yy);
        const float ly  = yy - y0f;
        const int   iy0 = (int)y0f;
        const int   iy1 = imin(iy0 + 1, HDIM - 1);
        const float* row0 = plane + iy0 * WDIM;
        const float* row1 = plane + iy1 * WDIM;
#pragma unroll
        for (int sx = 0; sx < 2; ++sx) {
            float xx = x1 + ((float)pw + (sx ? 0.75f : 0.25f)) * bin_w;
            xx = fminf(fmaxf(xx, 0.0f), 127.0f);
            const float x0f = floorf(xx);
            const float lx  = xx - x0f;
            const int   ix0 = (int)x0f;
            const int   ix1 = imin(ix0 + 1, WDIM - 1);

            const float v00 = row0[ix0];
            const float v01 = row0[ix1];
            const float v10 = row1[ix0];
            const float v11 = row1[ix1];

            const float hy = 1.0f - ly;
            const float hx = 1.0f - lx;
            acc += hy * hx * v00 + hy * lx * v01 + ly * hx * v10 + ly * lx * v11;
        }
    }
    out[idx] = acc * 0.25f; // mean over 2x2 samples
}

extern "C" void kernel_launch(void* const* d_in, const int* in_sizes, int n_in,
                              void* d_out, int out_size, void* d_ws, size_t ws_size,
                              hipStream_t stream) {
    const float* features = (const float*)d_in[0];
    const float* rois     = (const float*)d_in[1];
    float*       out      = (float*)d_out;

    const int R     = in_sizes[1] / 5;   // 2048
    const int total = out_size;          // R * 10 * 7 * 7

    const int block  = 256;
    const int blocks = (total + block - 1) / block;
    psroi_align_kernel<<<blocks, block, 0, stream>>>(features, rois, out, total, R);
}